// Admit_43207370998104
// MI455X (gfx1250) — compile-verified
//
#include <hip/hip_runtime.h>
#include <math.h>

typedef __attribute__((ext_vector_type(2))) float v2f;
typedef __attribute__((ext_vector_type(8))) float v8f;

#define B_      8
#define N_      2048
#define C_      512
#define CIN_    128
#define H_      96
#define W_      96
#define RECTS_  8
#define GROUPS_ 8

// d_out layout (float offsets): y, level1, level2, level3
#define OUT_L1_OFF  8388608   // 8*2048*512
#define OUT_L2_OFF  8519680   // +8*2048*2*4
#define OUT_L3_OFF  8781824   // +8*2048*4*4
// total out = 9306112 floats

// workspace layout (float offsets)
#define WS_RR_OFF   0          // 16384*8*4   final rects (unscaled)
#define WS_MASK_OFF 524288     // 16384*64    softmaxed mask
#define WS_S_OFF    1572864    // 131072*128  sampled S

// ---------------------------------------------------------------------------
// Stage A: ratios + mask logits GEMV, sigmoid/softmax, rect split tree.
// One block per (b,n) token row.
// ---------------------------------------------------------------------------
__global__ __launch_bounds__(128) void stageA(
    const float* __restrict__ x, const float* __restrict__ r,
    const float* __restrict__ Wr, const float* __restrict__ br,
    const float* __restrict__ Wm, const float* __restrict__ bm,
    float* __restrict__ out, float* __restrict__ rrws, float* __restrict__ maskws)
{
    __shared__ float xs[C_];
    __shared__ float logits[72];   // [0..6]=sigmoid ratios, [7..70]=mask logits
    const int bn  = blockIdx.x;    // 0 .. 16383  (= b*2048 + n)
    const int tid = threadIdx.x;

    const float* xr = x + (size_t)bn * C_;
    for (int i = tid; i < C_; i += 128) xs[i] = xr[i];
    __syncthreads();

    if (tid < 71) {
        float acc;
        if (tid < 7) {
            acc = br[tid];
            for (int c = 0; c < C_; ++c) acc += xs[c] * Wr[c * 7 + tid];
            logits[tid] = 1.f / (1.f + expf(-acc));         // sigmoid ratio
        } else {
            const int j = tid - 7;
            acc = bm[j];
            for (int c = 0; c < C_; ++c) acc += xs[c] * Wm[c * 64 + j];
            logits[tid] = acc;                               // raw mask logit
        }
    }
    __syncthreads();

    // softmax over rects (axis=2): per group g, over logits[s*8+g]
    if (tid < GROUPS_) {
        const int g = tid;
        float m = -INFINITY;
        for (int s = 0; s < RECTS_; ++s) m = fmaxf(m, logits[7 + s * 8 + g]);
        float e[RECTS_]; float sum = 0.f;
        for (int s = 0; s < RECTS_; ++s) { e[s] = expf(logits[7 + s * 8 + g] - m); sum += e[s]; }
        const float inv = 1.f / sum;
        for (int s = 0; s < RECTS_; ++s)
            maskws[((size_t)bn * RECTS_ + s) * GROUPS_ + g] = e[s] * inv;
    }

    // rect split tree (7 splits) — thread 0
    if (tid == 0) {
        float rr[8][4], nrr[8][4];
        const float* rp = r + (size_t)bn * 4;
        rr[0][0] = rp[0]; rr[0][1] = rp[1]; rr[0][2] = rp[2]; rr[0][3] = rp[3];
        const size_t offL[3] = { OUT_L1_OFF + (size_t)bn * 8,
                                 OUT_L2_OFF + (size_t)bn * 16,
                                 OUT_L3_OFF + (size_t)bn * 32 };
        int cnt = 1;
        for (int lev = 0; lev < 3; ++lev) {
            const int dim = lev & 1;                    // START_DIM=0: 0,1,0
            for (int j = 0; j < cnt; ++j) {
                const float rat = logits[(1 << lev) - 1 + j];
                const float mid = rr[j][dim] * (1.f - rat) + rr[j][dim + 2] * rat;
                for (int k = 0; k < 4; ++k) { nrr[j][k] = rr[j][k]; nrr[cnt + j][k] = rr[j][k]; }
                nrr[j][dim + 2]   = mid;                // r1 list first
                nrr[cnt + j][dim] = mid;                // then r2 list
            }
            cnt <<= 1;
            for (int j = 0; j < cnt; ++j)
                for (int k = 0; k < 4; ++k) rr[j][k] = nrr[j][k];
            float* lo = out + offL[lev];
            for (int j = 0; j < cnt; ++j)
                for (int k = 0; k < 4; ++k) lo[j * 4 + k] = rr[j][k];
        }
        float* rw = rrws + (size_t)bn * 32;
        for (int j = 0; j < 8; ++j)
            for (int k = 0; k < 4; ++k) rw[j * 4 + k] = rr[j][k];
    }
}

// ---------------------------------------------------------------------------
// Stage B: integral-image corner sampling. One block per rect, lane=channel.
// (X,Y) math is wave-uniform; 52 L2 gathers per thread. Images (151 MB) are
// L2-resident on MI455X (192 MB L2).
// ---------------------------------------------------------------------------
__global__ __launch_bounds__(128) void stageB(
    const float* __restrict__ I, const float* __restrict__ IX,
    const float* __restrict__ IY, const float* __restrict__ IT,
    const float* __restrict__ rrws, float* __restrict__ Sws)
{
    const int ridx = blockIdx.x;            // (b*2048+n)*8 + rect
    const int c    = threadIdx.x;           // channel
    const int b    = ridx >> 14;            // /(2048*8)
    const float sc = (float)(H_ - 1);       // H-1 == W-1 == 95

    const float x1 = rrws[(size_t)ridx * 4 + 0] * sc;
    const float y1 = rrws[(size_t)ridx * 4 + 1] * sc;
    const float x2 = rrws[(size_t)ridx * 4 + 2] * sc;
    const float y2 = rrws[(size_t)ridx * 4 + 3] * sc;

    const size_t base = ((size_t)(b * CIN_ + c)) * (H_ * W_);
    const float* Ib  = I  + base;
    const float* IXb = IX + base;
    const float* IYb = IY + base;
    const float* ITb = IT + base;

    auto tl = [&](float xx, float yy) -> float {
        int X = (int)ceilf(xx);
        int Y = (int)ceilf(yy);
        int Xm = X - 1; if (Xm < 0) Xm = 0;
        int Ym = Y - 1; if (Ym < 0) Ym = 0;
        const float dx = (float)X - xx, dy = (float)Y - yy;
        const float wx1 = 0.5f * dx * dx, wx2 = dx - wx1;
        const float wy1 = 0.5f * dy * dy, wy2 = dy - wy1;
        const int iXY = X * W_ + Y, iXYm = X * W_ + Ym;
        const int iXmY = Xm * W_ + Y, iXmYm = Xm * W_ + Ym;
        const float I_XY = Ib[iXY], I_XYm = Ib[iXYm];
        const float I_XmY = Ib[iXmY], I_XmYm = Ib[iXmYm];
        const float I_0Ym = Ib[Ym], I_0Y = Ib[Y];
        const float I_Xm0 = Ib[Xm * W_], I_X0 = Ib[X * W_];
        const float s_a  = ITb[iXY];
        const float s_e1 = wy1 * (IXb[iXYm] - 0.5f * I_XYm - 0.5f * I_0Ym)
                         + wy2 * (IXb[iXY]  - 0.5f * I_XY  - 0.5f * I_0Y);
        const float s_e2 = wx1 * (IYb[iXmY] - 0.5f * I_XmY - 0.5f * I_Xm0)
                         + wx2 * (IYb[iXY]  - 0.5f * I_XY  - 0.5f * I_X0);
        const float s_c  = wx1 * wy1 * I_XmYm + wx2 * wy1 * I_XYm
                         + wx1 * wy2 * I_XmY  + wx2 * wy2 * I_XY;
        return s_a - s_e1 - s_e2 + s_c;
    };

    const float S = tl(x2, y2) - tl(x1, y2) - tl(x2, y1) + tl(x1, y1);
    const float area = fabsf((x1 - x2) * (y1 - y2)) + 1e-9f;
    Sws[(size_t)ridx * CIN_ + c] = S / area;
}

// ---------------------------------------------------------------------------
// Stage C: S @ W_samp via V_WMMA_F32_16X16X4_F32 (exact f32), fused with
// mask-weighted rect combine, bias, LayerNorm, residual.
// Block = 8 waves, handles 4 tokens (= 32 S rows); each wave owns 4 column
// tiles of 16 channels; 256 WMMAs per wave.
// D layout: VGPR j, lane l -> row j + 8*(l/16), col l%16  => rect == j.
// ---------------------------------------------------------------------------
__global__ __launch_bounds__(256) void stageC(
    const float* __restrict__ Sws, const float* __restrict__ maskws,
    const float* __restrict__ Wsamp, const float* __restrict__ bsamp,
    const float* __restrict__ lng, const float* __restrict__ lnb,
    const float* __restrict__ x, float* __restrict__ out)
{
    __shared__ float sA[32 * CIN_];    // 32 S rows (4 tokens x 8 rects)
    __shared__ float yBuf[4 * C_];
    __shared__ float maskS[4 * 64];
    __shared__ float redS[256];
    __shared__ float redQ[256];

    const int tid = threadIdx.x;
    const int nb0 = blockIdx.x * 4;    // flat token base (b*N+n)

    {
        const float* src = Sws + (size_t)nb0 * RECTS_ * CIN_;
        for (int i = tid; i < 32 * CIN_; i += 256) sA[i] = src[i];
        maskS[tid] = maskws[(size_t)nb0 * 64 + tid];
    }
    __syncthreads();

    const int wave  = tid >> 5;
    const int lane  = tid & 31;
    const int lhalf = lane >> 4;   // 0/1: which K pair (A) and row half (D)
    const int l16   = lane & 15;

    for (int t = 0; t < 4; ++t) {
        const int nt  = wave * 4 + t;       // column tile 0..31
        const int col = nt * 16 + l16;      // output channel
        v8f acc0 = {};                      // rows 0..15  (tokens 0,1)
        v8f acc1 = {};                      // rows 16..31 (tokens 2,3)
        for (int k0 = 0; k0 < CIN_; k0 += 4) {
            const int k = k0 + 2 * lhalf;
            v2f bf, a0, a1;
            bf.x = Wsamp[(size_t)k * C_ + col];
            bf.y = Wsamp[(size_t)(k + 1) * C_ + col];
            a0.x = sA[l16 * CIN_ + k];
            a0.y = sA[l16 * CIN_ + k + 1];
            a1.x = sA[(16 + l16) * CIN_ + k];
            a1.y = sA[(16 + l16) * CIN_ + k + 1];
            acc0 = __builtin_amdgcn_wmma_f32_16x16x4_f32(
                false, a0, false, bf, (short)0, acc0, false, false);
            acc1 = __builtin_amdgcn_wmma_f32_16x16x4_f32(
                false, a1, false, bf, (short)0, acc1, false, false);
        }
        // rect-weighted combine: y[token][col] = sum_j mask[token][j][g]*d[j] + b
        const int g = col >> 6;
        const float bs = bsamp[col];
        {
            const int nlocal = lhalf;                 // tokens 0/1
            float yv = 0.f;
            #pragma unroll
            for (int j = 0; j < 8; ++j) yv += maskS[nlocal * 64 + j * 8 + g] * acc0[j];
            yBuf[nlocal * C_ + col] = yv + bs;
        }
        {
            const int nlocal = 2 + lhalf;             // tokens 2/3
            float yv = 0.f;
            #pragma unroll
            for (int j = 0; j < 8; ++j) yv += maskS[nlocal * 64 + j * 8 + g] * acc1[j];
            yBuf[nlocal * C_ + col] = yv + bs;
        }
    }
    __syncthreads();

    // LayerNorm + residual: 4 rows x 64 threads, 8 channels per thread
    const int row = tid >> 6;
    const int tt  = tid & 63;
    float sum = 0.f, sq = 0.f;
    #pragma unroll
    for (int u = 0; u < 8; ++u) {
        const float v = yBuf[row * C_ + tt * 8 + u];
        sum += v; sq += v * v;
    }
    redS[tid] = sum; redQ[tid] = sq;
    __syncthreads();
    for (int s = 32; s > 0; s >>= 1) {
        if (tt < s) { redS[tid] += redS[tid + s]; redQ[tid] += redQ[tid + s]; }
        __syncthreads();
    }
    const float mu   = redS[row * 64] * (1.f / 512.f);
    const float var  = redQ[row * 64] * (1.f / 512.f) - mu * mu;
    const float rstd = rsqrtf(var + 1e-5f);
    const float* xr = x   + (size_t)(nb0 + row) * C_;
    float*       yo = out + (size_t)(nb0 + row) * C_;
    #pragma unroll
    for (int u = 0; u < 8; ++u) {
        const int cc = tt * 8 + u;
        yo[cc] = (yBuf[row * C_ + cc] - mu) * rstd * lng[cc] + lnb[cc] + xr[cc];
    }
}

// ---------------------------------------------------------------------------
extern "C" void kernel_launch(void* const* d_in, const int* in_sizes, int n_in,
                              void* d_out, int out_size, void* d_ws, size_t ws_size,
                              hipStream_t stream)
{
    const float* x  = (const float*)d_in[0];
    const float* r  = (const float*)d_in[1];
    const float* I  = (const float*)d_in[2];
    const float* IX = (const float*)d_in[3];
    const float* IY = (const float*)d_in[4];
    const float* IT = (const float*)d_in[5];
    const float* Wr = (const float*)d_in[6];
    const float* br = (const float*)d_in[7];
    const float* Wm = (const float*)d_in[8];
    const float* bm = (const float*)d_in[9];
    const float* Wsamp = (const float*)d_in[10];
    const float* bsamp = (const float*)d_in[11];
    const float* lng = (const float*)d_in[12];
    const float* lnb = (const float*)d_in[13];
    float* out = (float*)d_out;
    float* ws  = (float*)d_ws;

    float* rrws   = ws + WS_RR_OFF;
    float* maskws = ws + WS_MASK_OFF;
    float* Sws    = ws + WS_S_OFF;

    stageA<<<B_ * N_,          128, 0, stream>>>(x, r, Wr, br, Wm, bm, out, rrws, maskws);
    stageB<<<B_ * N_ * RECTS_, 128, 0, stream>>>(I, IX, IY, IT, rrws, Sws);
    stageC<<<(B_ * N_) / 4,    256, 0, stream>>>(Sws, maskws, Wsamp, bsamp, lng, lnb, x, out);

    (void)in_sizes; (void)n_in; (void)out_size; (void)ws_size;
}